// Plastic_RetroactiveModulated_RNN_60163901882592
// MI455X (gfx1250) — compile-verified
//
#include <hip/hip_runtime.h>
#include <hip/hip_bf16.h>

// Problem dims (compile-time constants from the reference)
#define BB   128
#define INW  64
#define HS   500
#define HS4  (HS / 4)     // 125 float4 per row
#define NOUT 10

typedef float v2f __attribute__((ext_vector_type(2)));
typedef float v4f __attribute__((ext_vector_type(4)));
typedef float v8f __attribute__((ext_vector_type(8)));

// ---------------------------------------------------------------------------
// Kernel 1: pre[b,h] = sum_k inputs[b,k] * i2h_w[h,k] + i2h_b[h]
// via V_WMMA_F32_16X16X4_F32. One wave per 16x16 output tile.
//   grid = (8, 32)  [M tiles = 128/16, N tiles = ceil(500/16)]
// A fragment (16x4 f32): lanes 0-15 hold M=lane, VGPR0=K0 VGPR1=K1;
//                        lanes 16-31 hold M=lane-16, VGPR0=K2 VGPR1=K3.
// B fragment (4x16 f32): mirrored — lane&15 = N, half selects K pair.
// C/D (16x16 f32): VGPR v holds row M = v + 8*(lane>=16), col N = lane&15.
// OOB N columns: B row index is CLAMPED (branch-free); garbage only lands in
// D columns we never store, since D[:,n] depends only on B[:,n].
// ---------------------------------------------------------------------------
__global__ __launch_bounds__(32) void k_pre_wmma(const float* __restrict__ inputs,
                                                 const float* __restrict__ i2h_w,
                                                 const float* __restrict__ i2h_b,
                                                 float* __restrict__ pre) {
    const int lane = threadIdx.x & 31;
    const int mt   = blockIdx.x;       // 0..7
    const int nt   = blockIdx.y;       // 0..31
    const int half = lane >> 4;        // 0 or 1
    const int l15  = lane & 15;
    const int m    = mt * 16 + l15;    // A-row carried by this lane (<128 always)
    const int n    = nt * 16 + l15;    // B-col carried by this lane (may be >=500)
    const bool nok = (n < HS);
    const int  nc  = nok ? n : (HS - 1);   // clamped, always in-bounds

    const float* arow = inputs + (size_t)m * INW + 2 * half;  // a[v] = arow[k+v]
    const float* brow = i2h_w + (size_t)nc * INW + 2 * half;  // b[v] = brow[k+v]

    v8f c = {};
#pragma unroll
    for (int k = 0; k < INW; k += 4) {
        v2f a, bf;
        a.x  = arow[k];
        a.y  = arow[k + 1];
        bf.x = brow[k];
        bf.y = brow[k + 1];
        c = __builtin_amdgcn_wmma_f32_16x16x4_f32(
                /*neg_a=*/false, a, /*neg_b=*/false, bf,
                /*c_mod=*/(short)0, c, /*reuse_a=*/false, /*reuse_b=*/false);
    }

    if (nok) {
        const float bias = i2h_b[n];
#pragma unroll
        for (int v = 0; v < 8; ++v) {
            const int row = mt * 16 + v + 8 * half;   // < 128
            pre[(size_t)row * HS + n] = c[v] + bias;
        }
    }
}

// ---------------------------------------------------------------------------
// Kernel 2: rec[b,i] = sum_j (w[i,j] + alpha[i,j]*pw[b,i,j]) * hidden[b,j]
//           hactiv[b,i] = tanh(pre[b,i] + rec[b,i])   (written to hidden_new)
// One wave32 per (b,i) row; v4f streaming, shuffle reduction.
// First full pass over pw (128 MB) — loaded with DEFAULT (RT) temporal hint
// so the lines stay resident in the 192 MB L2 for the second pass.
// ---------------------------------------------------------------------------
__global__ __launch_bounds__(256) void k_rec(const float* __restrict__ hidden,
                                             const float* __restrict__ pw,
                                             const float* __restrict__ w,
                                             const float* __restrict__ alpha,
                                             const float* __restrict__ pre,
                                             float* __restrict__ hactiv) {
    const int wave = blockIdx.x * (blockDim.x >> 5) + (threadIdx.x >> 5);
    const int lane = threadIdx.x & 31;
    if (wave >= BB * HS) return;
    const int b = wave / HS;
    const int i = wave - b * HS;

    const v4f* pw4 = (const v4f*)(pw + (size_t)wave * HS);   // (b*HS+i)*HS
    const v4f* w4  = (const v4f*)(w  + (size_t)i * HS);
    const v4f* a4  = (const v4f*)(alpha + (size_t)i * HS);
    const v4f* h4  = (const v4f*)(hidden + (size_t)b * HS);

    float acc = 0.0f;
    for (int j = lane; j < HS4; j += 32) {
        const v4f pv = pw4[j];
        const v4f wv = w4[j];
        const v4f av = a4[j];
        const v4f hv = h4[j];
        acc += (wv.x + av.x * pv.x) * hv.x;
        acc += (wv.y + av.y * pv.y) * hv.y;
        acc += (wv.z + av.z * pv.z) * hv.z;
        acc += (wv.w + av.w * pv.w) * hv.w;
    }
#pragma unroll
    for (int off = 16; off > 0; off >>= 1) acc += __shfl_xor(acc, off, 32);
    if (lane == 0) hactiv[wave] = tanhf(pre[wave] + acc);
}

// ---------------------------------------------------------------------------
// Kernel 3: output heads. One block per batch, 12 waves = heads
//   head 0      : DAout[b]   = tanh(hactiv . h2DA_w + h2DA_b)
//   head 1..10  : activout   = hactiv . h2o_w[o]    + h2o_b[o]
//   head 11     : valueout   = hactiv . h2v_w       + h2v_b
// ---------------------------------------------------------------------------
__global__ __launch_bounds__(384) void k_heads(const float* __restrict__ hactiv,
                                               const float* __restrict__ h2DA_w,
                                               const float* __restrict__ h2DA_b,
                                               const float* __restrict__ h2o_w,
                                               const float* __restrict__ h2o_b,
                                               const float* __restrict__ h2v_w,
                                               const float* __restrict__ h2v_b,
                                               float* __restrict__ activout,
                                               float* __restrict__ valueout,
                                               float* __restrict__ DAout) {
    const int bidx = blockIdx.x;            // batch
    const int head = threadIdx.x >> 5;      // 0..11
    const int lane = threadIdx.x & 31;

    const float* hrow = hactiv + (size_t)bidx * HS;
    const float* wrow;
    if (head == 0)          wrow = h2DA_w;
    else if (head <= NOUT)  wrow = h2o_w + (size_t)(head - 1) * HS;
    else                    wrow = h2v_w;

    float acc = 0.0f;
    for (int j = lane; j < HS; j += 32) acc += hrow[j] * wrow[j];
#pragma unroll
    for (int off = 16; off > 0; off >>= 1) acc += __shfl_xor(acc, off, 32);

    if (lane == 0) {
        if (head == 0)         DAout[bidx] = tanhf(acc + h2DA_b[0]);
        else if (head <= NOUT) activout[(size_t)bidx * NOUT + head - 1] = acc + h2o_b[head - 1];
        else                   valueout[bidx] = acc + h2v_b[0];
    }
}

// ---------------------------------------------------------------------------
// Kernel 4: the dominant streaming pass over [B,HS,HS]:
//   pw_new   = clip(pw + DAout[b]*et, -1, 1)
//   et_new   = (1-etaet)*et + etaet * hactiv[b,i]*hidden[b,j]
//   hebb_new = hebb
// One block per (b,i) row; 125 active lanes each move one v4f of each tensor
// (B128 loads/stores).  Cache policy:
//   - pw   : regular (RT) load -> chance of L2 hit from k_rec's pass
//   - et/hebb loads and ALL stores: non-temporal (TH=NT) -> don't evict pw
// ---------------------------------------------------------------------------
__global__ __launch_bounds__(128) void k_update(const float* __restrict__ hebb,
                                                const float* __restrict__ et,
                                                const float* __restrict__ pw,
                                                const float* __restrict__ hidden,
                                                const float* __restrict__ hactiv,
                                                const float* __restrict__ DAout,
                                                const float* __restrict__ etaet_p,
                                                float* __restrict__ hebb_o,
                                                float* __restrict__ et_o,
                                                float* __restrict__ pw_o) {
    const int row = blockIdx.x;             // b*HS + i, 0..63999
    const int j4  = threadIdx.x;            // 0..127, active < 125
    if (j4 >= HS4) return;

    const int   b     = row / HS;
    const float da    = DAout[b];
    const float ha    = hactiv[row];
    const float etaet = etaet_p[0];
    const float om    = 1.0f - etaet;

    const size_t idx = (size_t)row * HS4 + j4;
    const v4f e  = __builtin_nontemporal_load((const v4f*)et + idx);
    const v4f hb = __builtin_nontemporal_load((const v4f*)hebb + idx);
    const v4f p  = ((const v4f*)pw)[idx];   // RT: may hit L2 from pass 1
    const v4f hv = ((const v4f*)hidden)[(size_t)b * HS4 + j4];

    v4f pn, en;
    pn.x = fminf(1.0f, fmaxf(-1.0f, fmaf(da, e.x, p.x)));
    pn.y = fminf(1.0f, fmaxf(-1.0f, fmaf(da, e.y, p.y)));
    pn.z = fminf(1.0f, fmaxf(-1.0f, fmaf(da, e.z, p.z)));
    pn.w = fminf(1.0f, fmaxf(-1.0f, fmaf(da, e.w, p.w)));

    en.x = fmaf(om, e.x, etaet * (ha * hv.x));
    en.y = fmaf(om, e.y, etaet * (ha * hv.y));
    en.z = fmaf(om, e.z, etaet * (ha * hv.z));
    en.w = fmaf(om, e.w, etaet * (ha * hv.w));

    __builtin_nontemporal_store(hb, (v4f*)hebb_o + idx);
    __builtin_nontemporal_store(en, (v4f*)et_o + idx);
    __builtin_nontemporal_store(pn, (v4f*)pw_o + idx);
}

// ---------------------------------------------------------------------------
// Host-side launcher
// ---------------------------------------------------------------------------
extern "C" void kernel_launch(void* const* d_in, const int* in_sizes, int n_in,
                              void* d_out, int out_size, void* d_ws, size_t ws_size,
                              hipStream_t stream) {
    // Inputs in setup_inputs() dict order:
    const float* inputs = (const float*)d_in[0];    // [128,64]
    const float* hidden = (const float*)d_in[1];    // [128,500]
    const float* hebb   = (const float*)d_in[2];    // [128,500,500]
    const float* et     = (const float*)d_in[3];    // [128,500,500]
    const float* pw     = (const float*)d_in[4];    // [128,500,500]
    const float* i2h_w  = (const float*)d_in[5];    // [500,64]
    const float* i2h_b  = (const float*)d_in[6];    // [500]
    const float* w      = (const float*)d_in[7];    // [500,500]
    const float* alpha  = (const float*)d_in[8];    // [500,500]
    // d_in[9] = eta (unused in the fully-modulated path)
    const float* etaet  = (const float*)d_in[10];   // [1]
    const float* h2DA_w = (const float*)d_in[11];   // [1,500]
    const float* h2DA_b = (const float*)d_in[12];   // [1]
    const float* h2o_w  = (const float*)d_in[13];   // [10,500]
    const float* h2o_b  = (const float*)d_in[14];   // [10]
    const float* h2v_w  = (const float*)d_in[15];   // [1,500]
    const float* h2v_b  = (const float*)d_in[16];   // [1]

    // Output layout (flat, return order):
    //   activout[128*10] | valueout[128] | hidden_new[128*500]
    //   | hebb_new[32M]  | et_new[32M]   | pw_new[32M]
    float* out       = (float*)d_out;
    float* activout  = out;
    float* valueout  = out + (size_t)BB * NOUT;                 // 1280
    float* hidden_o  = valueout + BB;                           // 1408
    float* hebb_o    = hidden_o + (size_t)BB * HS;              // 65408
    float* et_o      = hebb_o + (size_t)BB * HS * HS;           // +32M
    float* pw_o      = et_o   + (size_t)BB * HS * HS;           // +32M

    // Workspace: pre [128*500] + DAout [128]
    float* pre   = (float*)d_ws;
    float* DAout = pre + (size_t)BB * HS;

    // 1) pre = inputs @ i2h_w^T + b   (WMMA f32 16x16x4)
    k_pre_wmma<<<dim3(BB / 16, (HS + 15) / 16), dim3(32), 0, stream>>>(
        inputs, i2h_w, i2h_b, pre);

    // 2) rec + hactiv  (one wave per (b,i); 64000 waves, 8 waves/block)
    {
        const int waves = BB * HS;
        const int wpb   = 8;
        k_rec<<<dim3((waves + wpb - 1) / wpb), dim3(wpb * 32), 0, stream>>>(
            hidden, pw, w, alpha, pre, hidden_o);
    }

    // 3) heads (DAout / activout / valueout)
    k_heads<<<dim3(BB), dim3(12 * 32), 0, stream>>>(
        hidden_o, h2DA_w, h2DA_b, h2o_w, h2o_b, h2v_w, h2v_b,
        activout, valueout, DAout);

    // 4) streaming plasticity update (dominant pass)
    k_update<<<dim3(BB * HS), dim3(128), 0, stream>>>(
        hebb, et, pw, hidden, hidden_o, DAout, etaet,
        hebb_o, et_o, pw_o);
}